// GKernel_34806414966868
// MI455X (gfx1250) — compile-verified
//
#include <hip/hip_runtime.h>
#include <math.h>

// ---------------------------------------------------------------------------
// Graph-WL kernel similarity (N=20000, E=320000, L=16, F=16, M=32, D=128).
// Bottleneck = edge scatter (L2-resident, 192MB L2 holds both 32MB psi bufs).
// Dense math uses V_WMMA_F32_16X16X4_F32 (fp32-exact; FLOPs are negligible).
// Workspace layout (floats):
//   [0,               N*400)  Hd / psi ping buffer
//   [N*400,         2*N*400)  psi pong buffer (aliased as T scratch earlier)
//   [2*N*400, 2*N*400+F*400)  g  (filter readout)
//   [..., +F)                 gnorm
// Total ~64 MB.
// ---------------------------------------------------------------------------

typedef __attribute__((ext_vector_type(2))) float v2f;
typedef __attribute__((ext_vector_type(8))) float v8f;

// out[n*outStride + outOff + 4c .. +3] = in[n*inStride + inOff + 4c .. +3]
__global__ __launch_bounds__(256) void k_copy(
    const float* __restrict__ in, int inStride, int inOff,
    float* __restrict__ out, int outStride, int outOff,
    int W4, int N)
{
    int idx = blockIdx.x * 256 + threadIdx.x;
    if (idx >= N * W4) return;
    int n = idx / W4, c = idx % W4;
    const float4* s = reinterpret_cast<const float4*>(in + (size_t)n * inStride + inOff) + c;
    float4*       d = reinterpret_cast<float4*>(out + (size_t)n * outStride + outOff) + c;
    *d = *s;
}

// out[dst[e], :] += in[src[e], :]  (hardware GLOBAL_ATOMIC_ADD_F32, L2-resident)
__global__ __launch_bounds__(256) void k_scatter(
    const float* __restrict__ in, int inStride, int inOff,
    float* __restrict__ out, int outStride, int outOff,
    const int* __restrict__ srcI, const int* __restrict__ dstI,
    int E, int W4)
{
    int idx = blockIdx.x * 256 + threadIdx.x;
    if (idx >= E * W4) return;
    int e = idx / W4, c = idx % W4;        // consecutive lanes stream one edge row
    int s = srcI[e], d = dstI[e];
    const float4 v = *(reinterpret_cast<const float4*>(in + (size_t)s * inStride + inOff) + c);
    float* o = out + (size_t)d * outStride + outOff + (size_t)c * 4;
    unsafeAtomicAdd(o + 0, v.x);
    unsafeAtomicAdd(o + 1, v.y);
    unsafeAtomicAdd(o + 2, v.z);
    unsafeAtomicAdd(o + 3, v.w);
}

// out[:, outOff:outOff+128] = relu(A[N,K] @ W[K,128]) using fp32 WMMA 16x16x4.
// One wave per 16x16 tile; 8 waves/block cover the 128-wide output.
// f32 WMMA layouts (ISA 7.12.2): A 16x4: lane(l<16)->M=l, v0:K=0,v1:K=1;
// lanes 16-31 hold K=2,3.  B 4x16 mirrored.  C/D: lane -> N=l&15, vgpr v -> M=v+8*(l>>4).
__global__ __launch_bounds__(256) void k_gemm_relu(
    const float* __restrict__ A, int K,
    const float* __restrict__ W,          // [K,128]
    float* __restrict__ out, int outStride, int outOff, int N)
{
    int lane    = threadIdx.x & 31;
    int colTile = threadIdx.x >> 5;       // 0..7
    int rowTile = blockIdx.x;
    int r = lane & 15, half = lane >> 4;
    int row  = rowTile * 16 + r;
    int rowc = row < N ? row : N - 1;     // clamp (outputs of clamped rows are masked at store)
    int col  = colTile * 16 + r;
    const float* arow = A + (size_t)rowc * K;
    v8f acc = {0.f,0.f,0.f,0.f,0.f,0.f,0.f,0.f};
    for (int k0 = 0; k0 < K; k0 += 4) {
        int ka = k0 + 2 * half;
        v2f a, b;
        a.x = arow[ka];
        a.y = arow[ka + 1];
        b.x = W[(size_t)ka * 128 + col];
        b.y = W[(size_t)(ka + 1) * 128 + col];
        acc = __builtin_amdgcn_wmma_f32_16x16x4_f32(false, a, false, b,
                                                    (short)0, acc, false, false);
    }
#pragma unroll
    for (int v = 0; v < 8; ++v) {
        int m = rowTile * 16 + v + 8 * half;
        if (m < N) {
            float val = acc[v];
            out[(size_t)m * outStride + outOff + col] = val > 0.f ? val : 0.f;
        }
    }
}

// conv[N,16] = (psi @ g.T) / (||psi_row|| * ||g_row|| + eps), fp32 WMMA over K=400.
// Row norms computed inline from the A fragments; halves combined via shfl_xor(16).
__global__ __launch_bounds__(256) void k_sim(
    const float* __restrict__ psi,    // [N,400]
    const float* __restrict__ g,      // [16,400]
    const float* __restrict__ gnorm,  // [16]  (sum of squares)
    float* __restrict__ conv, int N)
{
    int wave   = threadIdx.x >> 5;
    int tile   = blockIdx.x * 8 + wave;
    int nTiles = (N + 15) >> 4;
    if (tile >= nTiles) return;       // wave-uniform -> scalar branch, WMMA never runs with EXEC!=all-1
    int lane = threadIdx.x & 31;
    int r = lane & 15, half = lane >> 4;
    int row  = tile * 16 + r;
    int rowc = row < N ? row : N - 1;
    const float* arow = psi + (size_t)rowc * 400;
    const float* brow = g   + (size_t)r * 400;   // column r of B == row r of g
    v8f acc = {0.f,0.f,0.f,0.f,0.f,0.f,0.f,0.f};
    float pn = 0.f;
    for (int k0 = 0; k0 < 400; k0 += 4) {
        int ka = k0 + 2 * half;
        v2f a, b;
        a.x = arow[ka]; a.y = arow[ka + 1];
        b.x = brow[ka]; b.y = brow[ka + 1];
        pn += a.x * a.x + a.y * a.y;
        acc = __builtin_amdgcn_wmma_f32_16x16x4_f32(false, a, false, b,
                                                    (short)0, acc, false, false);
    }
    pn += __shfl_xor(pn, 16, 32);     // lane l now has ||psi[tile*16 + (l&15)]||^2
    float gs = sqrtf(gnorm[r]);
#pragma unroll
    for (int v = 0; v < 8; ++v) {
        int mloc = v + 8 * half;
        int m = tile * 16 + mloc;
        float pm = __shfl(pn, mloc, 32);
        if (m < N)
            conv[(size_t)m * 16 + r] = acc[v] / (sqrtf(pm) * gs + 1e-8f);
    }
}

// Filter-graph WL pipeline + readout. One block per filter (F=16, M=32). Tiny.
__global__ __launch_bounds__(256) void k_filter(
    const float* __restrict__ P,  const float* __restrict__ X,
    const float* __restrict__ W0, const float* __restrict__ W1,
    const float* __restrict__ W2,
    float* __restrict__ g, float* __restrict__ gnorm)
{
    __shared__ float sX[32 * 16], sP[32 * 32], sT16[32 * 16];
    __shared__ float sH[32 * 128], sT[32 * 128], sg[400];
    int f = blockIdx.x, tid = threadIdx.x;
    for (int i = tid; i < 512;  i += 256) sX[i] = X[f * 512 + i];
    for (int i = tid; i < 1024; i += 256) sP[i] = P[f * 1024 + i];
    __syncthreads();
    if (tid < 16) {                       // g[0:16] = colsum(X)  (deterministic)
        float s = 0.f;
        for (int m = 0; m < 32; ++m) s += sX[m * 16 + tid];
        sg[tid] = s;
    }
    for (int i = tid; i < 512; i += 256) {   // T16 = X + P@X
        int m = i >> 4, l = i & 15;
        float acc = sX[i];
        for (int n = 0; n < 32; ++n) acc += sP[m * 32 + n] * sX[n * 16 + l];
        sT16[i] = acc;
    }
    __syncthreads();
    for (int i = tid; i < 4096; i += 256) {  // H1 = relu(T16 @ W0)
        int m = i >> 7, d = i & 127;
        float acc = 0.f;
        for (int l = 0; l < 16; ++l) acc += sT16[m * 16 + l] * W0[l * 128 + d];
        sH[i] = fmaxf(acc, 0.f);
    }
    __syncthreads();
    if (tid < 128) {                      // g[16:144] = colsum(H1)
        float s = 0.f;
        for (int m = 0; m < 32; ++m) s += sH[m * 128 + tid];
        sg[16 + tid] = s;
    }
    const float* Ws[2] = {W1, W2};
    const int    go[2] = {144, 272};
    for (int it = 0; it < 2; ++it) {
        __syncthreads();
        for (int i = tid; i < 4096; i += 256) {  // T = H + P@H
            int m = i >> 7, d = i & 127;
            float acc = sH[i];
            for (int n = 0; n < 32; ++n) acc += sP[m * 32 + n] * sH[n * 128 + d];
            sT[i] = acc;
        }
        __syncthreads();
        for (int i = tid; i < 4096; i += 256) {  // H = relu(T @ W)
            int m = i >> 7, d = i & 127;
            float acc = 0.f;
            const float* Wp = Ws[it];
            for (int k = 0; k < 128; ++k) acc += sT[m * 128 + k] * Wp[k * 128 + d];
            sH[i] = fmaxf(acc, 0.f);
        }
        __syncthreads();
        if (tid < 128) {
            float s = 0.f;
            for (int m = 0; m < 32; ++m) s += sH[m * 128 + tid];
            sg[go[it] + tid] = s;
        }
    }
    __syncthreads();
    for (int i = tid; i < 400; i += 256) g[f * 400 + i] = sg[i];
    if (tid == 0) {
        float s = 0.f;
        for (int i = 0; i < 400; ++i) s += sg[i] * sg[i];
        gnorm[f] = s;
    }
}

extern "C" void kernel_launch(void* const* d_in, const int* in_sizes, int n_in,
                              void* d_out, int out_size, void* d_ws, size_t ws_size,
                              hipStream_t stream)
{
    const float* x  = (const float*)d_in[0];   // [N,16]
    const int*   ei = (const int*)  d_in[1];   // [2,E]
    const float* P  = (const float*)d_in[2];   // [F,32,32]
    const float* X  = (const float*)d_in[3];   // [F,32,16]
    const float* W0 = (const float*)d_in[4];   // [16,128]
    const float* W1 = (const float*)d_in[5];   // [128,128]
    const float* W2 = (const float*)d_in[6];   // [128,128]

    const int L = 16, D = 128, FT = 400;       // FT = L + 3*D
    int N = in_sizes[0] / L;
    int E = in_sizes[1] / 2;
    int F = in_sizes[2] / (32 * 32);
    const int* srcI = ei;
    const int* dstI = ei + E;

    float* Hd  = (float*)d_ws;                 // psi ping  [N,400]
    float* ps2 = Hd + (size_t)N * FT;          // psi pong  [N,400]
    float* Tb  = ps2;                          // T scratch aliases pong (dead until hops)
    float* g   = ps2 + (size_t)N * FT;         // [F,400]
    float* gn  = g + (size_t)F * FT;           // [F]
    float* conv = (float*)d_out;

    auto cdiv = [](long long a, long long b) { return (unsigned)((a + b - 1) / b); };
    int nTiles = (N + 15) / 16;

    // Hd[:,0:16] = x   (WL round-0 features)
    k_copy<<<cdiv((long long)N * 4, 256), 256, 0, stream>>>(x, L, 0, Hd, FT, 0, 4, N);
    // T0 = x + agg(x)
    k_copy<<<cdiv((long long)N * 4, 256), 256, 0, stream>>>(x, L, 0, Tb, L, 0, 4, N);
    k_scatter<<<cdiv((long long)E * 4, 256), 256, 0, stream>>>(x, L, 0, Tb, L, 0, srcI, dstI, E, 4);
    // H1 = relu(T0 @ W0) -> Hd[:,16:144]
    k_gemm_relu<<<nTiles, 256, 0, stream>>>(Tb, L, W0, Hd, FT, 16, N);
    // T1 = H1 + agg(H1)
    k_copy<<<cdiv((long long)N * 32, 256), 256, 0, stream>>>(Hd, FT, 16, Tb, D, 0, 32, N);
    k_scatter<<<cdiv((long long)E * 32, 256), 256, 0, stream>>>(Hd, FT, 16, Tb, D, 0, srcI, dstI, E, 32);
    // H2 = relu(T1 @ W1) -> Hd[:,144:272]
    k_gemm_relu<<<nTiles, 256, 0, stream>>>(Tb, D, W1, Hd, FT, 144, N);
    // T2 = H2 + agg(H2)
    k_copy<<<cdiv((long long)N * 32, 256), 256, 0, stream>>>(Hd, FT, 144, Tb, D, 0, 32, N);
    k_scatter<<<cdiv((long long)E * 32, 256), 256, 0, stream>>>(Hd, FT, 144, Tb, D, 0, srcI, dstI, E, 32);
    // H3 = relu(T2 @ W2) -> Hd[:,272:400]
    k_gemm_relu<<<nTiles, 256, 0, stream>>>(Tb, D, W2, Hd, FT, 272, N);

    // 3-hop egonet pooling: psi <- psi + agg(psi), ping-pong Hd <-> ps2
    k_copy<<<cdiv((long long)N * 100, 256), 256, 0, stream>>>(Hd, FT, 0, ps2, FT, 0, 100, N);
    k_scatter<<<cdiv((long long)E * 100, 256), 256, 0, stream>>>(Hd, FT, 0, ps2, FT, 0, srcI, dstI, E, 100);
    k_copy<<<cdiv((long long)N * 100, 256), 256, 0, stream>>>(ps2, FT, 0, Hd, FT, 0, 100, N);
    k_scatter<<<cdiv((long long)E * 100, 256), 256, 0, stream>>>(ps2, FT, 0, Hd, FT, 0, srcI, dstI, E, 100);
    k_copy<<<cdiv((long long)N * 100, 256), 256, 0, stream>>>(Hd, FT, 0, ps2, FT, 0, 100, N);
    k_scatter<<<cdiv((long long)E * 100, 256), 256, 0, stream>>>(Hd, FT, 0, ps2, FT, 0, srcI, dstI, E, 100);

    // filter-graph WL + readout (tiny)
    k_filter<<<F, 256, 0, stream>>>(P, X, W0, W1, W2, g, gn);

    // normalized similarity, WMMA over K=400, psi = ps2 after 3 hops
    k_sim<<<cdiv(nTiles, 8), 256, 0, stream>>>(ps2, g, gn, conv, N);
}